// DGI_19181323944511
// MI455X (gfx1250) — compile-verified
//
#include <hip/hip_runtime.h>
#include <hip/hip_bf16.h>

#define DF 128
#define BN_EPS 1e-5f

typedef float v2f __attribute__((ext_vector_type(2)));
typedef float v8f __attribute__((ext_vector_type(8)));

// ---------------------------------------------------------------------------
// Fused GEMM: out[r, :] = A[rowmap(r), :] @ W1 + (HAS_M ? M[r, :] @ W2 : 0) + bias
// optionally ReLU. D = 128. Uses V_WMMA_F32_16X16X4_F32.
// 256 threads = 8 waves; wave w owns rows [blk*128 + w*16, +16), all 128 cols.
// Weights staged in LDS pair-interleaved so each B fragment is ONE ds_load_b64
// into an even-aligned VGPR pair:
//   element (k, n) of the 32-row chunk lives at lds[((k>>1)*128 + n)*2 + (k&1)]
// Per k-step, ALL 8 B fragments are fetched into distinct registers before the
// WMMA chain so LDS latency overlaps the previous step's matrix ops.
// ---------------------------------------------------------------------------
template <bool HAS_M, bool RELU>
__global__ __launch_bounds__(256)
void dgi_gemm_wmma(const float* __restrict__ A,
                   const int*   __restrict__ rowmap,   // optional row indirection for A
                   const float* __restrict__ M,        // second operand (mean) if HAS_M
                   const float* __restrict__ W1,       // [128,128] row-major (k, n)
                   const float* __restrict__ W2,       // [128,128] if HAS_M
                   const float* __restrict__ bias,     // [128]
                   float*       __restrict__ out,      // [nrows,128]
                   int nrows)
{
    __shared__ float lw1[32 * DF];
    __shared__ float lw2[HAS_M ? 32 * DF : 1];
    __shared__ float lbias[DF];

    const int tid  = threadIdx.x;
    const int wave = tid >> 5;
    const int lane = tid & 31;
    const int half = lane >> 4;   // K-pair select for A/B tiles
    const int lm   = lane & 15;   // M (for A) / N (for B,C,D) index

    const int rbase = blockIdx.x * 128 + wave * 16;

    int rowA = rbase + lm;
    if (rowA >= nrows) rowA = nrows - 1;          // clamp (stores are masked)
    const int arow = rowmap ? rowmap[rowA] : rowA;

    if (tid < DF) lbias[tid] = bias[tid];

    const float* Abase = A + (long)arow * DF + 2 * half;
    const float* Mbase = HAS_M ? (M + (long)rowA * DF + 2 * half) : nullptr;

    v8f acc[8];
#pragma unroll
    for (int i = 0; i < 8; ++i) acc[i] = (v8f)0.0f;

    for (int kc = 0; kc < DF; kc += 32) {
        __syncthreads();
        // stage 32 rows of W1 (and W2) into LDS, pair-interleaved.
        // pair p = (kp, n): kp = p>>7 in [0,16), n = p&127; one v2f store each.
#pragma unroll
        for (int i = 0; i < 8; ++i) {
            const int p  = tid + i * 256;
            const int kp = p >> 7;
            const int n  = p & 127;
            v2f w;
            w.x = W1[(size_t)(kc + 2 * kp) * DF + n];
            w.y = W1[(size_t)(kc + 2 * kp + 1) * DF + n];
            *(v2f*)(lw1 + (size_t)p * 2) = w;
            if (HAS_M) {
                v2f m2;
                m2.x = W2[(size_t)(kc + 2 * kp) * DF + n];
                m2.y = W2[(size_t)(kc + 2 * kp + 1) * DF + n];
                *(v2f*)(lw2 + (size_t)p * 2) = m2;
            }
        }
        __syncthreads();

        // distance-1 software pipeline on the A fragments (8B-aligned b64 loads)
        v2f a1c = *(const v2f*)(Abase + kc);
        v2f a2c = (v2f)0.0f;
        if (HAS_M) a2c = *(const v2f*)(Mbase + kc);

#pragma unroll
        for (int q = 0; q < 8; ++q) {
            v2f a1n = a1c, a2n = a2c;
            if (q < 7) {
                a1n = *(const v2f*)(Abase + kc + (q + 1) * 4);
                if (HAS_M) a2n = *(const v2f*)(Mbase + kc + (q + 1) * 4);
            }
            // fetch ALL B fragments for this k-step into distinct registers
            v2f B1[8];
            v2f B2[8];
#pragma unroll
            for (int nt = 0; nt < 8; ++nt) {
                const int fp = (2 * q + half) * DF + nt * 16 + lm;
                B1[nt] = *(const v2f*)(lw1 + (size_t)fp * 2);
                if (HAS_M) B2[nt] = *(const v2f*)(lw2 + (size_t)fp * 2);
            }
#pragma unroll
            for (int nt = 0; nt < 8; ++nt) {
                acc[nt] = __builtin_amdgcn_wmma_f32_16x16x4_f32(
                    false, a1c, false, B1[nt], (short)0, acc[nt], false, false);
                if (HAS_M) {
                    acc[nt] = __builtin_amdgcn_wmma_f32_16x16x4_f32(
                        false, a2c, false, B2[nt], (short)0, acc[nt], false, false);
                }
            }
            a1c = a1n;
            a2c = a2n;
        }
    }

    // epilogue: bias (+ReLU) and masked stores. C/D layout: VGPR j -> row j + 8*half
#pragma unroll
    for (int nt = 0; nt < 8; ++nt) {
        const int col = nt * 16 + lm;
        const float bv = lbias[col];
#pragma unroll
        for (int j = 0; j < 8; ++j) {
            const int row = rbase + j + 8 * half;
            if (row < nrows) {
                float v = acc[nt][j] + bv;
                if (RELU) v = fmaxf(v, 0.0f);
                out[(long)row * DF + col] = v;
            }
        }
    }
}

// ---------------------------------------------------------------------------
// Edge scatter-add: agg[dst] += h[perm?(src)], deg[dst] += 1.  One thread per
// (edge, 16B chunk): float4 gather + 4 fp32 global atomics.
// ---------------------------------------------------------------------------
__global__ __launch_bounds__(256)
void dgi_scatter_add(const float* __restrict__ h,
                     const int* __restrict__ src,
                     const int* __restrict__ dst,
                     const int* __restrict__ perm,
                     float* __restrict__ agg,
                     float* __restrict__ deg,
                     int nedges)
{
    const long t = (long)blockIdx.x * blockDim.x + threadIdx.x;
    const long e = t >> 5;
    const int  c = (int)(t & 31);
    if (e >= (long)nedges) return;
    int s = src[e];
    if (perm) s = perm[s];
    const int d = dst[e];
    const float4 v = ((const float4*)(h + (long)s * DF))[c];
    float* ap = agg + (long)d * DF + c * 4;
    atomicAdd(ap + 0, v.x);
    atomicAdd(ap + 1, v.y);
    atomicAdd(ap + 2, v.z);
    atomicAdd(ap + 3, v.w);
    if (c == 0) atomicAdd(deg + d, 1.0f);
}

// agg[r,:] *= 1 / max(deg[r], 1)
__global__ __launch_bounds__(256)
void dgi_mean(float* __restrict__ agg, const float* __restrict__ deg, int n)
{
    const long t = (long)blockIdx.x * blockDim.x + threadIdx.x;
    const long r = t >> 5;
    const int  c = (int)(t & 31);
    if (r >= (long)n) return;
    const float inv = 1.0f / fmaxf(deg[r], 1.0f);
    float4* p = (float4*)(agg + r * DF) + c;
    float4 v = *p;
    v.x *= inv; v.y *= inv; v.z *= inv; v.w *= inv;
    *p = v;
}

// per-column sum & sum-of-squares (atomic accumulation into 128-wide buffers)
__global__ __launch_bounds__(256)
void dgi_colstat(const float* __restrict__ h, int n,
                 float* __restrict__ sums, float* __restrict__ sumsq)
{
    __shared__ float ssum[256];
    __shared__ float ssq[256];
    const int col = threadIdx.x & 127;
    const int sub = threadIdx.x >> 7;   // 0 or 1
    float s = 0.0f, q = 0.0f;
    for (long r = (long)blockIdx.x * 2 + sub; r < (long)n; r += (long)gridDim.x * 2) {
        const float v = h[r * DF + col];
        s += v; q += v * v;
    }
    ssum[threadIdx.x] = s;
    ssq[threadIdx.x]  = q;
    __syncthreads();
    if (sub == 0) {
        atomicAdd(&sums[col],  ssum[col] + ssum[col + 128]);
        atomicAdd(&sumsq[col], ssq[col]  + ssq[col + 128]);
    }
}

// BatchNorm (population var) + ReLU, in place
__global__ __launch_bounds__(256)
void dgi_bn_relu(float* __restrict__ h, int n,
                 const float* __restrict__ sums, const float* __restrict__ sumsq,
                 const float* __restrict__ gamma, const float* __restrict__ beta)
{
    const long t = (long)blockIdx.x * blockDim.x + threadIdx.x;
    const long r = t >> 5;
    const int  c = (int)(t & 31);
    if (r >= (long)n) return;
    const float inv_n = 1.0f / (float)n;
    float4* p = (float4*)(h + r * DF) + c;
    float4 v = *p;
    float o[4] = {v.x, v.y, v.z, v.w};
#pragma unroll
    for (int i = 0; i < 4; ++i) {
        const int col = c * 4 + i;
        const float m   = sums[col] * inv_n;
        const float var = sumsq[col] * inv_n - m * m;
        o[i] = fmaxf(gamma[col] * (o[i] - m) * rsqrtf(var + BN_EPS) + beta[col], 0.0f);
    }
    v.x = o[0]; v.y = o[1]; v.z = o[2]; v.w = o[3];
    *p = v;
}

// summary = sigmoid(colmean(positive)); ws = disc_W @ summary
__global__ __launch_bounds__(128)
void dgi_ws(const float* __restrict__ discW, const float* __restrict__ sums,
            float inv_n, float* __restrict__ wsv)
{
    __shared__ float summ[DF];
    const int i = threadIdx.x;
    summ[i] = 1.0f / (1.0f + expf(-(sums[i] * inv_n)));
    __syncthreads();
    float acc = 0.0f;
    for (int j = 0; j < DF; ++j) acc += discW[i * DF + j] * summ[j];
    wsv[i] = acc;
}

// mean(softplus(sign * (feat @ ws))) accumulated into *loss (scale = 1/n)
__global__ __launch_bounds__(256)
void dgi_disc_loss(const float* __restrict__ feat, int n,
                   const float* __restrict__ wsv, float sign, float scale,
                   float* __restrict__ loss)
{
    __shared__ float lws[DF];
    __shared__ float red[256];
    if (threadIdx.x < DF) lws[threadIdx.x] = wsv[threadIdx.x];
    __syncthreads();
    const long r = (long)blockIdx.x * blockDim.x + threadIdx.x;
    float val = 0.0f;
    if (r < (long)n) {
        float acc = 0.0f;
        const float* row = feat + r * DF;
        for (int j = 0; j < DF; j += 4)
            acc += row[j] * lws[j] + row[j + 1] * lws[j + 1]
                 + row[j + 2] * lws[j + 2] + row[j + 3] * lws[j + 3];
        const float x = sign * acc;
        val = (fmaxf(x, 0.0f) + log1pf(expf(-fabsf(x)))) * scale;  // stable softplus
    }
    red[threadIdx.x] = val;
    __syncthreads();
    for (int s = 128; s > 0; s >>= 1) {
        if (threadIdx.x < s) red[threadIdx.x] += red[threadIdx.x + s];
        __syncthreads();
    }
    if (threadIdx.x == 0) atomicAdd(loss, red[0]);
}

// z[e,:] = feat[a[e],:] * feat[b[e],:]
__global__ __launch_bounds__(256)
void dgi_pair_mul(const float* __restrict__ feat, const int* __restrict__ a,
                  const int* __restrict__ b, float* __restrict__ z, int n)
{
    const long t = (long)blockIdx.x * blockDim.x + threadIdx.x;
    const long e = t >> 5;
    const int  c = (int)(t & 31);
    if (e >= (long)n) return;
    const float4 va = ((const float4*)(feat + (long)a[e] * DF))[c];
    const float4 vb = ((const float4*)(feat + (long)b[e] * DF))[c];
    float4 o;
    o.x = va.x * vb.x; o.y = va.y * vb.y; o.z = va.z * vb.z; o.w = va.w * vb.w;
    ((float4*)(z + e * DF))[c] = o;
}

// out[e] = z[e,:] @ w + b[0]
__global__ __launch_bounds__(256)
void dgi_mlp_out(const float* __restrict__ z, const float* __restrict__ w,
                 const float* __restrict__ b, float* __restrict__ out, int n)
{
    __shared__ float lw[DF];
    if (threadIdx.x < DF) lw[threadIdx.x] = w[threadIdx.x];
    __syncthreads();
    const long r = (long)blockIdx.x * blockDim.x + threadIdx.x;
    if (r >= (long)n) return;
    float acc = b[0];
    const float* row = z + r * DF;
    for (int j = 0; j < DF; ++j) acc += row[j] * lw[j];
    out[r] = acc;
}

// ---------------------------------------------------------------------------
extern "C" void kernel_launch(void* const* d_in, const int* in_sizes, int n_in,
                              void* d_out, int out_size, void* d_ws, size_t ws_size,
                              hipStream_t stream)
{
    (void)in_sizes; (void)n_in; (void)out_size; (void)ws_size;

    const int N0 = 100000, N1 = 50000, N2 = 25000, N3 = 12500;
    const int E0 = 750000, E1 = 250000, E2 = 125000, EP = 10000;
    (void)N0;

    const float* x    = (const float*)d_in[0];
    const int* src0   = (const int*)d_in[1];
    const int* dst0   = (const int*)d_in[2];
    const int* src1   = (const int*)d_in[3];
    const int* dst1   = (const int*)d_in[4];
    const int* src2   = (const int*)d_in[5];
    const int* dst2   = (const int*)d_in[6];
    const int* perm   = (const int*)d_in[7];
    const int* pos_s  = (const int*)d_in[8];
    const int* pos_d  = (const int*)d_in[9];
    const int* neg_s  = (const int*)d_in[10];
    const int* neg_d  = (const int*)d_in[11];
    const float* Wself0  = (const float*)d_in[12];
    const float* Wneigh0 = (const float*)d_in[13];
    const float* b0      = (const float*)d_in[14];
    const float* Wself1  = (const float*)d_in[15];
    const float* Wneigh1 = (const float*)d_in[16];
    const float* b1      = (const float*)d_in[17];
    const float* Wself2  = (const float*)d_in[18];
    const float* Wneigh2 = (const float*)d_in[19];
    const float* b2      = (const float*)d_in[20];
    const float* gamma0  = (const float*)d_in[21];
    const float* beta0   = (const float*)d_in[22];
    const float* gamma1  = (const float*)d_in[23];
    const float* beta1   = (const float*)d_in[24];
    const float* disc_W  = (const float*)d_in[25];
    const float* pW1 = (const float*)d_in[26];
    const float* pb1 = (const float*)d_in[27];
    const float* pW2 = (const float*)d_in[28];
    const float* pb2 = (const float*)d_in[29];
    const float* pW3 = (const float*)d_in[30];
    const float* pb3 = (const float*)d_in[31];

    // --- scratch carve-out -------------------------------------------------
    char* sp = (char*)d_ws;
    size_t off = 0;
    auto carve = [&](size_t bytes) -> void* {
        void* p = sp + off;
        off += (bytes + 255) & ~(size_t)255;
        return p;
    };
    float* agg   = (float*)carve((size_t)N1 * DF * 4);
    float* h1    = (float*)carve((size_t)N1 * DF * 4);
    float* h2    = (float*)carve((size_t)N2 * DF * 4);
    float* posf  = (float*)carve((size_t)N3 * DF * 4);
    float* negf  = (float*)carve((size_t)N3 * DF * 4);
    float* degb  = (float*)carve((size_t)N1 * 4);
    float* sums  = (float*)carve(DF * 4);
    float* sumsq = (float*)carve(DF * 4);
    float* wsv   = (float*)carve(DF * 4);
    float* z1    = (float*)carve((size_t)EP * DF * 4);
    float* z2    = (float*)carve((size_t)EP * DF * 4);

    auto run_layer = [&](const float* hin, const int* rmap,
                         const int* s, const int* d, int ndst, int nedges,
                         const float* Ws, const float* Wn, const float* bias,
                         float* hout) {
        hipMemsetAsync(agg, 0, (size_t)ndst * DF * 4, stream);
        hipMemsetAsync(degb, 0, (size_t)ndst * 4, stream);
        const long st = (long)nedges * 32;
        dgi_scatter_add<<<(int)((st + 255) / 256), 256, 0, stream>>>(
            hin, s, d, rmap, agg, degb, nedges);
        const long mt = (long)ndst * 32;
        dgi_mean<<<(int)((mt + 255) / 256), 256, 0, stream>>>(agg, degb, ndst);
        dgi_gemm_wmma<true, false><<<(ndst + 127) / 128, 256, 0, stream>>>(
            hin, rmap, agg, Ws, Wn, bias, hout, ndst);
    };
    auto run_bn = [&](float* h, int n, const float* g, const float* be) {
        hipMemsetAsync(sums, 0, DF * 4, stream);
        hipMemsetAsync(sumsq, 0, DF * 4, stream);
        dgi_colstat<<<128, 256, 0, stream>>>(h, n, sums, sumsq);
        const long t = (long)n * 32;
        dgi_bn_relu<<<(int)((t + 255) / 256), 256, 0, stream>>>(h, n, sums, sumsq, g, be);
    };
    auto encode = [&](const int* rmap, float* outf) {
        run_layer(x,  rmap,    src0, dst0, N1, E0, Wself0, Wneigh0, b0, h1);
        run_bn(h1, N1, gamma0, beta0);
        run_layer(h1, nullptr, src1, dst1, N2, E1, Wself1, Wneigh1, b1, h2);
        run_bn(h2, N2, gamma1, beta1);
        run_layer(h2, nullptr, src2, dst2, N3, E2, Wself2, Wneigh2, b2, outf);
    };

    encode(nullptr, posf);   // positive: raw features
    encode(perm,    negf);   // negative: permuted features (via index indirection)

    // --- discriminator loss ------------------------------------------------
    float* out = (float*)d_out;
    hipMemsetAsync(sums, 0, DF * 4, stream);
    hipMemsetAsync(sumsq, 0, DF * 4, stream);
    dgi_colstat<<<128, 256, 0, stream>>>(posf, N3, sums, sumsq);
    dgi_ws<<<1, 128, 0, stream>>>(disc_W, sums, 1.0f / (float)N3, wsv);
    float* loss_ptr = out + 20000;
    hipMemsetAsync(loss_ptr, 0, 4, stream);
    dgi_disc_loss<<<(N3 + 255) / 256, 256, 0, stream>>>(
        posf, N3, wsv, -1.0f, 1.0f / (float)N3, loss_ptr);
    dgi_disc_loss<<<(N3 + 255) / 256, 256, 0, stream>>>(
        negf, N3, wsv, +1.0f, 1.0f / (float)N3, loss_ptr);

    // --- edge predictor (both heads use POSITIVE embeddings) --------------
    const long pt = (long)EP * 32;
    // positive pairs -> out[0..EP)
    dgi_pair_mul<<<(int)((pt + 255) / 256), 256, 0, stream>>>(posf, pos_s, pos_d, z1, EP);
    dgi_gemm_wmma<false, true><<<(EP + 127) / 128, 256, 0, stream>>>(
        z1, nullptr, nullptr, pW1, nullptr, pb1, z2, EP);
    dgi_gemm_wmma<false, true><<<(EP + 127) / 128, 256, 0, stream>>>(
        z2, nullptr, nullptr, pW2, nullptr, pb2, z1, EP);
    dgi_mlp_out<<<(EP + 255) / 256, 256, 0, stream>>>(z1, pW3, pb3, out, EP);
    // negative pairs -> out[EP..2*EP)
    dgi_pair_mul<<<(int)((pt + 255) / 256), 256, 0, stream>>>(posf, neg_s, neg_d, z1, EP);
    dgi_gemm_wmma<false, true><<<(EP + 127) / 128, 256, 0, stream>>>(
        z1, nullptr, nullptr, pW1, nullptr, pb1, z2, EP);
    dgi_gemm_wmma<false, true><<<(EP + 127) / 128, 256, 0, stream>>>(
        z2, nullptr, nullptr, pW2, nullptr, pb2, z1, EP);
    dgi_mlp_out<<<(EP + 255) / 256, 256, 0, stream>>>(z1, pW3, pb3, out + EP, EP);
}